// LaplacianLoss_10960756539821
// MI455X (gfx1250) — compile-verified
//
#include <hip/hip_runtime.h>
#include <stdint.h>

typedef float v2f __attribute__((ext_vector_type(2)));
typedef float v8f __attribute__((ext_vector_type(8)));

constexpr int NV     = 128 * 128;   // 16384 vertices (2^14)
constexpr int NB     = 8;           // batch
constexpr int NCOL   = 32;          // columns padded (24 real = 8 batches x 3 dims)
constexpr int KC     = 64;          // K-chunk size
constexpr int LP     = 68;          // padded LDS row stride in dwords (16B aligned)
constexpr int WAVES  = 4;           // waves per workgroup (wave32)
constexpr int NCHUNK = NV / KC;     // 256

// ---- CDNA5 async global->LDS copy (ASYNCcnt-tracked), GVS addressing ----
__device__ __forceinline__ void async_copy_b128(uint32_t lds_off, uint32_t goff,
                                                const float* base)
{
    asm volatile("global_load_async_to_lds_b128 %0, %1, %2"
                 :: "v"(lds_off), "v"(goff), "s"(base)
                 : "memory");
}
__device__ __forceinline__ void wait_async0()
{
    asm volatile("s_wait_asynccnt 0x0" ::: "memory");
}

// Build Xg[n][k] = x[n/3, k, n%3] for n<24, 0 for n in [24,32): [32][NV] in d_ws.
__global__ __launch_bounds__(256)
void xg_prep(const float* __restrict__ x, float* __restrict__ Xg)
{
    const int idx = blockIdx.x * 256 + threadIdx.x;   // 0 .. 32*NV
    const int n   = idx >> 14;                        // NV == 2^14
    const int k   = idx & (NV - 1);
    float v = 0.0f;
    if (n < 24)
        v = x[(size_t)(n / 3) * NV * 3 + (size_t)k * 3 + (n % 3)];
    Xg[idx] = v;
}

// Y = L @ X (X = Xg[n][k] transposed view). Each wave owns a 16-row strip of L,
// accumulating a 16x32 tile of Y over K with V_WMMA_F32_16X16X4_F32.
// Double-buffered async global->LDS staging overlaps the 1 GiB L stream with math.
__global__ __launch_bounds__(128)
void lap_wmma_kernel(const float* __restrict__ L,
                     const float* __restrict__ Xg,
                     float* __restrict__ partials)
{
    __shared__ float Lsh[2][WAVES][16][LP];   // 34.8 KB
    __shared__ float Xsh[2][NCOL][LP];        // 17.4 KB
    __shared__ float sacc[NB];

    const int tid  = threadIdx.x;
    const int wave = tid >> 5;
    const int lane = tid & 31;
    const int row0 = blockIdx.x * (WAVES * 16) + wave * 16;

    if (tid < NB) sacc[tid] = 0.0f;

    // Issue one K-chunk of async copies (12 instr/wave): this wave's 16xKC strip
    // of L, plus 8 rows of the shared X chunk.
    auto issue_chunk = [&](int k0, int buf) {
        const uint32_t kbyte = (uint32_t)k0 * 4u;
        #pragma unroll
        for (int i = 0; i < 8; ++i) {
            const int item = i * 32 + lane;            // 256 float4 tiles
            const int r    = item >> 4;                // 0..15
            const int c4   = (item & 15) << 2;         // float col within chunk
            const uint32_t goff = ((uint32_t)(row0 + r) * (uint32_t)NV + c4) * 4u + kbyte;
            async_copy_b128((uint32_t)(uintptr_t)&Lsh[buf][wave][r][c4], goff, L);
        }
        #pragma unroll
        for (int j = 0; j < 4; ++j) {
            const int item = j * 32 + lane;            // 128 float4 tiles
            const int nr   = wave * 8 + (item >> 4);   // this wave's 8 X rows
            const int c4   = (item & 15) << 2;
            const uint32_t goff = ((uint32_t)nr * (uint32_t)NV + c4) * 4u + kbyte;
            async_copy_b128((uint32_t)(uintptr_t)&Xsh[buf][nr][c4], goff, Xg);
        }
    };

    v8f acc0 = {};   // cols 0..15
    v8f acc1 = {};   // cols 16..31 (16..23 meaningful)

    // A(16x4)/B(4x16) f32 fragments: lane l, vgpr v -> (l%16, 2*(l/16)+v)
    // => one 8B-aligned ds_load_b64 per fragment from [row][k] LDS layouts.
    const int frow = lane & 15;
    const int fk   = (lane >> 4) << 1;

    issue_chunk(0, 0);

    for (int it = 0; it < NCHUNK; ++it) {
        const int buf = it & 1;
        wait_async0();        // own async copies for chunk `it` complete
        __syncthreads();      // all waves' copies visible; prev buffer free
        if (it + 1 < NCHUNK) issue_chunk((it + 1) * KC, buf ^ 1);

        const float* lfrag  = &Lsh[buf][wave][frow][fk];
        const float* xfrag0 = &Xsh[buf][frow][fk];
        const float* xfrag1 = &Xsh[buf][16 + frow][fk];
        #pragma unroll
        for (int kk = 0; kk < KC; kk += 4) {
            const v2f a  = *(const v2f*)(lfrag  + kk);
            const v2f b0 = *(const v2f*)(xfrag0 + kk);
            const v2f b1 = *(const v2f*)(xfrag1 + kk);
            acc0 = __builtin_amdgcn_wmma_f32_16x16x4_f32(
                       false, a, false, b0, (short)0, acc0, false, false);
            acc1 = __builtin_amdgcn_wmma_f32_16x16x4_f32(
                       false, a, false, b1, (short)0, acc1, false, false);
        }
    }

    // Epilogue: each lane owns a fixed output column; sum y^2 per batch.
    float s0 = 0.0f, s1 = 0.0f;
    #pragma unroll
    for (int r = 0; r < 8; ++r) {
        s0 += acc0[r] * acc0[r];
        s1 += acc1[r] * acc1[r];
    }
    const int n0 = lane & 15;                 // acc0 column (< 24 always)
    atomicAdd(&sacc[n0 / 3], s0);
    if (n0 < 8)                               // acc1 column 16+n0 < 24
        atomicAdd(&sacc[(16 + n0) / 3], s1);
    __syncthreads();

    if (tid < NB) partials[blockIdx.x * NB + tid] = sacc[tid];
}

// Deterministic final reduction: out[b] = (sum of partials) / (nv*3)
__global__ void lap_reduce_kernel(const float* __restrict__ partials,
                                  float* __restrict__ out, int nwg)
{
    const int b = threadIdx.x;
    if (b < NB) {
        float s = 0.0f;
        for (int w = 0; w < nwg; ++w) s += partials[w * NB + b];
        out[b] = s * (1.0f / ((float)NV * 3.0f));
    }
}

extern "C" void kernel_launch(void* const* d_in, const int* in_sizes, int n_in,
                              void* d_out, int out_size, void* d_ws, size_t ws_size,
                              hipStream_t stream)
{
    const float* L = (const float*)d_in[0];            // [NV*NV] f32
    const float* x = (const float*)d_in[1];            // [NB, NV, 3] f32
    float* Xg       = (float*)d_ws;                    // [32][NV] = 2 MB scratch
    float* partials = Xg + (size_t)NCOL * NV;          // nwg*NB floats
    const int nwg = NV / (WAVES * 16);                 // 256 workgroups

    xg_prep<<<(NCOL * NV) / 256, 256, 0, stream>>>(x, Xg);
    lap_wmma_kernel<<<nwg, 128, 0, stream>>>(L, Xg, partials);
    lap_reduce_kernel<<<1, 32, 0, stream>>>(partials, (float*)d_out, nwg);
}